// XLinearMultiHeadedAttention_28827820491302
// MI455X (gfx1250) — compile-verified
//
#include <hip/hip_runtime.h>
#include <hip/hip_bf16.h>
#include <math.h>

// ---------------------------------------------------------------------------
// XLinearMultiHeadedAttention for MI455X (gfx1250, wave32, WMMA f32 16x16x4)
//
// Sizes: B=16, Q=K=100, H=8, DM=512, DK=64, MID=32
// Workspace layout (floats):
//   proj  : 4 * H * 1600 * 64   (p: 0=q1, 1=k, 2=q2, 3=v), layout [p][h][b*100+l][d]
//   alpha : B*H*100*100
//   ch    : B*H*100*64
// ---------------------------------------------------------------------------

typedef __attribute__((ext_vector_type(2))) float v2f;
typedef __attribute__((ext_vector_type(8))) float v8f;

#define NB    16
#define NL    100
#define NH    8
#define NDM   512
#define NDK   64
#define NMID  32
#define NROW  1600          // NB * NL
#define PROJ_FLOATS   (4u * NH * NROW * NDK)       // 3,276,800
#define ALPHA_FLOATS  ((size_t)NB * NH * NL * NL)  // 1,280,000
#define CH_FLOATS     ((size_t)NB * NH * NL * NDK) // 819,200

__device__ __forceinline__ v8f wmma_f32(v2f a, v2f b, v8f c) {
  // D = A(16x4,f32) x B(4x16,f32) + C(16x16,f32)
  return __builtin_amdgcn_wmma_f32_16x16x4_f32(
      /*neg_a=*/false, a, /*neg_b=*/false, b,
      /*c_mod=*/(short)0, c, /*reuse_a=*/false, /*reuse_b=*/false);
}

__device__ __forceinline__ float celu1(float x) {
  return x > 0.0f ? x : expm1f(x);
}

// ---------------------------------------------------------------------------
// Kernel 1: four fused projections: y = GN(CELU(x @ W^T + b)) * gw + gb
// One wave computes a 16-row x 64-col (= one head group) tile so GroupNorm
// reduces entirely inside the wave via shfl_xor over 16-lane halves.
// grid: 4 proj * 8 heads * 100 row-tiles = 3200 waves = 400 blocks of 256.
// ---------------------------------------------------------------------------
__global__ __launch_bounds__(256) void k_proj(
    const float* __restrict__ query, const float* __restrict__ key,
    const float* __restrict__ value,
    const float* __restrict__ Wq1, const float* __restrict__ bq1,
    const float* __restrict__ gw1, const float* __restrict__ gb1,
    const float* __restrict__ Wk,  const float* __restrict__ bk,
    const float* __restrict__ gwk, const float* __restrict__ gbk,
    const float* __restrict__ Wq2, const float* __restrict__ bq2,
    const float* __restrict__ gw2, const float* __restrict__ gb2,
    const float* __restrict__ Wv,  const float* __restrict__ bv,
    const float* __restrict__ gwv, const float* __restrict__ gbv,
    float* __restrict__ proj)
{
  const int lane = threadIdx.x & 31;
  const int wave = (blockIdx.x * blockDim.x + threadIdx.x) >> 5;
  if (wave >= 4 * NH * 100) return;

  const int p       = wave / (NH * 100);
  const int rem     = wave % (NH * 100);
  const int h       = rem / 100;
  const int rowTile = rem % 100;

  const float* x  = (p == 1) ? key : ((p == 3) ? value : query);
  const float* W  = (p == 0) ? Wq1 : (p == 1) ? Wk  : (p == 2) ? Wq2 : Wv;
  const float* bb = (p == 0) ? bq1 : (p == 1) ? bk  : (p == 2) ? bq2 : bv;
  const float* gw = (p == 0) ? gw1 : (p == 1) ? gwk : (p == 2) ? gw2 : gwv;
  const float* gb = (p == 0) ? gb1 : (p == 1) ? gbk : (p == 2) ? gb2 : gbv;

  const int half  = lane >> 4;        // 0: lanes 0-15, 1: lanes 16-31
  const int l16   = lane & 15;
  const int nbase = rowTile * 16;
  const int obase = h * NDK;
  const int arow  = nbase + l16;      // A-matrix row for this lane

  v8f acc[4] = {};
  for (int c = 0; c < NDM; c += 4) {
    const int ca = c + half * 2;
    v2f a;
    a.x = x[(size_t)arow * NDM + ca];
    a.y = x[(size_t)arow * NDM + ca + 1];
#pragma unroll
    for (int t = 0; t < 4; ++t) {
      const int o = obase + t * 16 + l16;
      v2f bf;
      bf.x = W[(size_t)o * NDM + ca];
      bf.y = W[(size_t)o * NDM + ca + 1];
      acc[t] = wmma_f32(a, bf, acc[t]);
    }
  }

  // Epilogue: bias + CELU, then GroupNorm over the 64 head channels per row.
  float y[4][8];
#pragma unroll
  for (int r = 0; r < 8; ++r) {
    float s1 = 0.0f, s2 = 0.0f;
#pragma unroll
    for (int t = 0; t < 4; ++t) {
      float yy = acc[t][r] + bb[obase + t * 16 + l16];
      yy = celu1(yy);
      y[t][r] = yy;
      s1 += yy;
      s2 += yy * yy;
    }
#pragma unroll
    for (int m = 1; m < 16; m <<= 1) {   // reduce within the 16-lane half
      s1 += __shfl_xor(s1, m, 32);
      s2 += __shfl_xor(s2, m, 32);
    }
    const float mu   = s1 * (1.0f / 64.0f);
    const float var  = s2 * (1.0f / 64.0f) - mu * mu;
    const float rstd = rsqrtf(var + 1e-5f);
    const int   row  = nbase + r + half * 8;    // C layout: M=r (lo) / r+8 (hi)
#pragma unroll
    for (int t = 0; t < 4; ++t) {
      const int o = obase + t * 16 + l16;
      const float outv = (y[t][r] - mu) * rstd * gw[o] + gb[o];
      proj[(((size_t)p * NH + h) * NROW + row) * NDK + (t * 16 + l16)] = outv;
    }
  }
}

// ---------------------------------------------------------------------------
// Kernel 2: key_map GEMM, fused relu/alpha/pool/squeeze.
//   km[q,k,m] = relu( sum_d q1[q,d] * (k[k,d]*We[m,d]) + be[m] )
//   alpha[q,k] = sum_m km*Ws[m] + bs     (pre-mask; masking done in softmax)
//   pool[q,m]  = sum_k mask*km / sum_k mask
//   ch[q,d]    = sigmoid(pool @ Wsq^T + bsq)
// One wave per (b,h,q-tile); grid = 16*8*7 = 896 blocks of 32.
// ---------------------------------------------------------------------------
__global__ __launch_bounds__(32) void k_keymap(
    const float* __restrict__ proj, const float* __restrict__ mask,
    const float* __restrict__ We,   const float* __restrict__ be,
    const float* __restrict__ Ws,   const float* __restrict__ bs,
    const float* __restrict__ Wsq,  const float* __restrict__ bsq,
    float* __restrict__ alpha, float* __restrict__ ch)
{
  __shared__ float WeT[NDK][NMID];    // [d][m] transposed -> conflict-free reads
  __shared__ float kkL[NL][NDK];      // key rows for this (b,h)
  __shared__ float maskL[16][NL];     // mask tile for this q-tile
  __shared__ float poolL[16][NMID];

  const int lane = threadIdx.x;
  const int blk  = blockIdx.x;
  const int qt   = blk % 7;
  const int bh   = blk / 7;
  const int h    = bh % NH;
  const int b    = bh / NH;
  const int qbase = qt * 16;
  const int half  = lane >> 4;
  const int l16   = lane & 15;

  const float* q1 = proj + (((size_t)0 * NH + h) * NROW + (size_t)b * NL) * NDK;
  const float* kk = proj + (((size_t)1 * NH + h) * NROW + (size_t)b * NL) * NDK;

  for (int i = lane; i < NMID * NDK; i += 32) {
    const int m = i >> 6, d = i & 63;
    WeT[d][m] = We[i];                       // We is (MID, DK)
  }
  for (int i = lane; i < NL * NDK; i += 32)
    kkL[i >> 6][i & 63] = kk[i];
  for (int i = lane; i < 16 * NL; i += 32) {
    const int r = i / NL, k2 = i % NL;
    const int q = qbase + r;
    maskL[r][k2] = (q < NL) ? mask[((size_t)b * NL + q) * NL + k2] : 0.0f;
  }
  __syncthreads();

  // Preload q1 A-fragments (16x64 tile -> 16 frags of K=4)
  const int  arow = qbase + l16;
  const bool arv  = arow < NL;
  v2f afr[16];
#pragma unroll
  for (int i = 0; i < 16; ++i) {
    const int dA = i * 4 + half * 2;
    afr[i].x = arv ? q1[(size_t)arow * NDK + dA]     : 0.0f;
    afr[i].y = arv ? q1[(size_t)arow * NDK + dA + 1] : 0.0f;
  }

  const float wsA = Ws[l16], wsB = Ws[16 + l16];
  const float beA = be[l16], beB = be[16 + l16];
  const float bs0 = bs[0];

  v8f pool0 = {}, pool1 = {};
  float msum8[8];
#pragma unroll
  for (int r = 0; r < 8; ++r) msum8[r] = 0.0f;

  float* alphaBase = alpha + (size_t)bh * NL * NL;   // [q][k]

  for (int k = 0; k < NL; ++k) {
    v8f c0 = {}, c1 = {};
#pragma unroll
    for (int i = 0; i < 16; ++i) {
      const int dA = i * 4 + half * 2;
      const float k0 = kkL[k][dA];
      const float k1 = kkL[k][dA + 1];
      v2f b0, b1;
      b0.x = k0 * WeT[dA][l16];        b0.y = k1 * WeT[dA + 1][l16];
      b1.x = k0 * WeT[dA][16 + l16];   b1.y = k1 * WeT[dA + 1][16 + l16];
      c0 = wmma_f32(afr[i], b0, c0);
      c1 = wmma_f32(afr[i], b1, c1);
    }
    float part[8];
#pragma unroll
    for (int r = 0; r < 8; ++r) {
      float x0 = fmaxf(c0[r] + beA, 0.0f);
      float x1 = fmaxf(c1[r] + beB, 0.0f);
      const float mrow = maskL[r + half * 8][k];
      pool0[r] += mrow * x0;
      pool1[r] += mrow * x1;
      msum8[r] += mrow * (1.0f / 16.0f);  // each of the 16 lanes adds; net = mrow
      part[r] = x0 * wsA + x1 * wsB;
    }
#pragma unroll
    for (int r = 0; r < 8; ++r) {
      float s = part[r];
#pragma unroll
      for (int m2 = 1; m2 < 16; m2 <<= 1) s += __shfl_xor(s, m2, 32);
      part[r] = s;
    }
    if (l16 == 0) {
#pragma unroll
      for (int r = 0; r < 8; ++r) {
        const int q = qbase + r + half * 8;
        if (q < NL) alphaBase[(size_t)q * NL + k] = part[r] + bs0;
      }
    }
  }

  // Normalize pool and stash to LDS (all 16 lanes of a half hold same msum).
#pragma unroll
  for (int r = 0; r < 8; ++r) {
    float msum = msum8[r] * 16.0f;
    const float inv = 1.0f / msum;
    poolL[r + half * 8][l16]      = pool0[r] * inv;
    poolL[r + half * 8][16 + l16] = pool1[r] * inv;
  }
  __syncthreads();

  // ch = sigmoid(pool @ Wsq^T + bsq) : 16 rows x 64 channels
  float* chp = ch + (size_t)bh * NL * NDK;
  for (int i = lane; i < 16 * NDK; i += 32) {
    const int r = i >> 6, d = i & 63;
    const int q = qbase + r;
    if (q < NL) {
      float s = bsq[d];
#pragma unroll
      for (int m = 0; m < NMID; ++m) s += poolL[r][m] * Wsq[d * NMID + m];
      chp[(size_t)q * NDK + d] = 1.0f / (1.0f + expf(-s));
    }
  }
}

// ---------------------------------------------------------------------------
// Kernel 3: masked softmax over k for each (b,h,q) row. One wave per row.
// ---------------------------------------------------------------------------
__global__ __launch_bounds__(256) void k_softmax(
    const float* __restrict__ mask, float* __restrict__ alpha)
{
  const int lane = threadIdx.x & 31;
  const int row  = (blockIdx.x * blockDim.x + threadIdx.x) >> 5;
  if (row >= NB * NH * NL) return;

  const int q  = row % NL;
  const int bh = row / NL;
  const int b  = bh / NH;

  float* a = alpha + (size_t)row * NL;
  const float* mrow = mask + ((size_t)b * NL + q) * NL;

  float v[4];
  float mx = -3.0e38f;
#pragma unroll
  for (int j = 0; j < 4; ++j) {
    const int k = lane + j * 32;
    if (k < NL) {
      float x = a[k];
      if (mrow[k] == 0.0f) x = -1.0e9f;
      v[j] = x;
      mx = fmaxf(mx, x);
    } else {
      v[j] = -3.0e38f;
    }
  }
#pragma unroll
  for (int m = 1; m < 32; m <<= 1) mx = fmaxf(mx, __shfl_xor(mx, m, 32));

  float s = 0.0f;
#pragma unroll
  for (int j = 0; j < 4; ++j) {
    const int k = lane + j * 32;
    if (k < NL) { v[j] = expf(v[j] - mx); s += v[j]; }
  }
#pragma unroll
  for (int m = 1; m < 32; m <<= 1) s += __shfl_xor(s, m, 32);

  const float inv = 1.0f / s;
#pragma unroll
  for (int j = 0; j < 4; ++j) {
    const int k = lane + j * 32;
    if (k < NL) a[k] = v[j] * inv;
  }
}

// ---------------------------------------------------------------------------
// Kernel 4: out = q2 * (alpha @ v) * ch, stored as (B, Q, H*DK).
// One wave per (b,h,q-tile) computes a 16x64 tile; K=100 = 25 WMMA K-steps.
// ---------------------------------------------------------------------------
__global__ __launch_bounds__(32) void k_attnout(
    const float* __restrict__ proj, const float* __restrict__ alpha,
    const float* __restrict__ ch, float* __restrict__ out)
{
  const int lane = threadIdx.x;
  const int blk  = blockIdx.x;
  const int qt   = blk % 7;
  const int bh   = blk / 7;
  const int h    = bh % NH;
  const int b    = bh / NH;
  const int qbase = qt * 16;
  const int half  = lane >> 4;
  const int l16   = lane & 15;

  const float* al = alpha + (size_t)bh * NL * NL;
  const float* vv = proj + (((size_t)3 * NH + h) * NROW + (size_t)b * NL) * NDK;

  const int  arow = qbase + l16;
  const bool arv  = arow < NL;

  v8f acc[4] = {};
  for (int k = 0; k < NL; k += 4) {
    const int kA = k + half * 2;
    v2f a;
    a.x = arv ? al[(size_t)arow * NL + kA]     : 0.0f;
    a.y = arv ? al[(size_t)arow * NL + kA + 1] : 0.0f;
#pragma unroll
    for (int t = 0; t < 4; ++t) {
      const int d = t * 16 + l16;
      v2f bf;
      bf.x = vv[(size_t)kA * NDK + d];
      bf.y = vv[(size_t)(kA + 1) * NDK + d];
      acc[t] = wmma_f32(a, bf, acc[t]);
    }
  }

  const float* q2  = proj + (((size_t)2 * NH + h) * NROW + (size_t)b * NL) * NDK;
  const float* chp = ch + (size_t)bh * NL * NDK;
#pragma unroll
  for (int r = 0; r < 8; ++r) {
    const int q = qbase + r + half * 8;
    if (q < NL) {
#pragma unroll
      for (int t = 0; t < 4; ++t) {
        const int d = t * 16 + l16;
        const float val =
            q2[(size_t)q * NDK + d] * acc[t][r] * chp[(size_t)q * NDK + d];
        out[((size_t)(b * NL + q)) * NDM + h * NDK + d] = val;
      }
    }
  }
}

// ---------------------------------------------------------------------------
extern "C" void kernel_launch(void* const* d_in, const int* in_sizes, int n_in,
                              void* d_out, int out_size, void* d_ws, size_t ws_size,
                              hipStream_t stream) {
  (void)in_sizes; (void)n_in; (void)out_size; (void)ws_size;

  const float* query = (const float*)d_in[0];
  const float* key   = (const float*)d_in[1];
  const float* value = (const float*)d_in[2];
  const float* mask  = (const float*)d_in[3];
  const float* Wq1 = (const float*)d_in[4];  const float* bq1 = (const float*)d_in[5];
  const float* gw1 = (const float*)d_in[6];  const float* gb1 = (const float*)d_in[7];
  const float* Wk  = (const float*)d_in[8];  const float* bk  = (const float*)d_in[9];
  const float* gwk = (const float*)d_in[10]; const float* gbk = (const float*)d_in[11];
  const float* Wq2 = (const float*)d_in[12]; const float* bq2 = (const float*)d_in[13];
  const float* gw2 = (const float*)d_in[14]; const float* gb2 = (const float*)d_in[15];
  const float* Wv  = (const float*)d_in[16]; const float* bv  = (const float*)d_in[17];
  const float* gwv = (const float*)d_in[18]; const float* gbv = (const float*)d_in[19];
  const float* We  = (const float*)d_in[20]; const float* be  = (const float*)d_in[21];
  const float* Ws  = (const float*)d_in[22]; const float* bs  = (const float*)d_in[23];
  const float* Wsq = (const float*)d_in[24]; const float* bsq = (const float*)d_in[25];

  float* ws    = (float*)d_ws;
  float* proj  = ws;
  float* alpha = ws + PROJ_FLOATS;
  float* ch    = alpha + ALPHA_FLOATS;
  float* outp  = (float*)d_out;

  // K1: 3200 wave-tiles / 8 waves per block = 400 blocks
  k_proj<<<400, 256, 0, stream>>>(query, key, value,
      Wq1, bq1, gw1, gb1, Wk, bk, gwk, gbk,
      Wq2, bq2, gw2, gb2, Wv, bv, gwv, gbv, proj);

  // K2: one 32-thread wave per (b,h,q-tile)
  k_keymap<<<NB * NH * 7, 32, 0, stream>>>(proj, mask, We, be, Ws, bs, Wsq, bsq,
                                           alpha, ch);

  // K3: 12800 rows * 32 lanes = 409600 threads
  k_softmax<<<(NB * NH * NL * 32 + 255) / 256, 256, 0, stream>>>(mask, alpha);

  // K4: one wave per (b,h,q-tile)
  k_attnout<<<NB * NH * 7, 32, 0, stream>>>(proj, alpha, ch, outp);
}